// IntraGraphAttention_20194936226353
// MI455X (gfx1250) — compile-verified
//
#include <hip/hip_runtime.h>
#include <math.h>

#define NN 50000
#define EE 1600000
#define DIN 128
#define HC  64   // H * C = 2 * 32

typedef float v2f __attribute__((ext_vector_type(2)));
typedef float v8f __attribute__((ext_vector_type(8)));

// Native CDNA5 f32 atomics (non-returning, device scope -> RMW at L2).
// Guarantees global_atomic_add_f32 / global_atomic_max_num_f32 instead of a
// possible cmpswap retry loop from the generic atomicAdd/atomicMax paths.
__device__ inline void atomicAddF32(float* p, float v) {
    asm volatile("global_atomic_add_f32 %0, %1, off scope:SCOPE_DEV"
                 :: "v"(p), "v"(v) : "memory");
}
__device__ inline void atomicMaxF32(float* p, float v) {
    asm volatile("global_atomic_max_num_f32 %0, %1, off scope:SCOPE_DEV"
                 :: "v"(p), "v"(v) : "memory");
}

// ---------------------------------------------------------------------------
// Kernel 1: fused ELU + 4 projections via V_WMMA_F32_16X16X4_F32 (fp32 exact).
// Block = 128 threads = 4 waves; wave w computes matrix w (Q/K/V/skip) for a
// 16-row tile. Skip result is written directly to d_out (residual init).
// ---------------------------------------------------------------------------
__global__ __launch_bounds__(128) void gemm_qkvs(
    const float* __restrict__ x,
    const float* __restrict__ Wq, const float* __restrict__ bq,
    const float* __restrict__ Wk, const float* __restrict__ bk,
    const float* __restrict__ Wv, const float* __restrict__ bv,
    const float* __restrict__ Ws, const float* __restrict__ bs,
    float* __restrict__ Q, float* __restrict__ K, float* __restrict__ V,
    float* __restrict__ out)
{
    __shared__ float xs[16 * 132];   // 16 rows x 128 cols, padded (132%64=4) vs bank conflicts

    const int tile = blockIdx.x;
    const int tid  = threadIdx.x;

    // Cooperative coalesced load of the 16x128 tile with fused ELU.
    #pragma unroll
    for (int j = 0; j < 16; ++j) {
        int idx = j * 128 + tid;              // 0..2047
        int row = idx >> 7, col = idx & 127;
        float v = x[(size_t)(tile * 16 + row) * DIN + col];
        xs[row * 132 + col] = v > 0.f ? v : expm1f(v);
    }
    __syncthreads();

    const int wave = tid >> 5;
    const int lane = tid & 31;

    const float* W; const float* bias; float* D;
    if      (wave == 0) { W = Wq; bias = bq; D = Q;   }
    else if (wave == 1) { W = Wk; bias = bk; D = K;   }
    else if (wave == 2) { W = Wv; bias = bv; D = V;   }
    else                { W = Ws; bias = bs; D = out; }

    const int mrow = lane & 15;           // M index (A) / N index (B)
    const int kb   = (lane >> 4) << 1;    // lanes 16-31 hold K=2,3

    const v8f z = {0.f, 0.f, 0.f, 0.f, 0.f, 0.f, 0.f, 0.f};
    v8f acc[4] = {z, z, z, z};

    for (int k0 = 0; k0 < DIN; k0 += 4) {
        // A fragment: 16x4 fp32, lane layout per ISA 7.12.2
        v2f a;
        a.x = xs[mrow * 132 + k0 + kb];
        a.y = xs[mrow * 132 + k0 + kb + 1];

        const float* w0 = W + (size_t)(k0 + kb) * HC + mrow;  // row-major [128x64]
        #pragma unroll
        for (int ct = 0; ct < 4; ++ct) {
            v2f b;
            b.x = w0[ct * 16];            // K = k0+kb
            b.y = w0[ct * 16 + HC];       // K = k0+kb+1
            acc[ct] = __builtin_amdgcn_wmma_f32_16x16x4_f32(
                false, a, false, b, (short)0, acc[ct], false, false);
        }
    }

    // C/D layout: VGPR r -> M = r + 8*(lane>=16), N = lane&15
    const int n     = lane & 15;
    const int rbase = tile * 16 + ((lane >> 4) << 3);
    #pragma unroll
    for (int ct = 0; ct < 4; ++ct) {
        int col  = ct * 16 + n;
        float bb = bias[col];
        #pragma unroll
        for (int r = 0; r < 8; ++r)
            D[(size_t)(rbase + r) * HC + col] = acc[ct][r] + bb;
    }
}

// ---------------------------------------------------------------------------
// Kernel 0: init segment-max (-FLT_MAX-ish) and denom (0).
// ---------------------------------------------------------------------------
__global__ void init_md(float* __restrict__ m, float* __restrict__ denom)
{
    int i = blockIdx.x * 256 + threadIdx.x;
    if (i < NN * 2) { m[i] = -3.0e38f; denom[i] = 0.f; }
}

// ---------------------------------------------------------------------------
// Kernel 2: per-edge logits + segment max. 16 lanes per edge, float4 slices,
// shfl_xor tree reduce within each 8-lane head group, native f32 max atomic.
// ---------------------------------------------------------------------------
__global__ __launch_bounds__(256) void edge_alpha(
    const int* __restrict__ ei, const float* __restrict__ Q,
    const float* __restrict__ K, float* __restrict__ alpha,
    float* __restrict__ mbuf)
{
    int t = blockIdx.x * 256 + threadIdx.x;
    int e = t >> 4;
    if (e >= EE) return;
    int l   = t & 15;
    int src = ei[e];
    int dst = ei[EE + e];

    const float4 qa = ((const float4*)(Q + (size_t)dst * HC))[l];
    const float4 ka = ((const float4*)(K + (size_t)src * HC))[l];
    float p = qa.x * ka.x + qa.y * ka.y + qa.z * ka.z + qa.w * ka.w;
    p += __shfl_xor(p, 1);
    p += __shfl_xor(p, 2);
    p += __shfl_xor(p, 4);      // reduced across the 8 lanes of one head

    if ((l & 7) == 0) {
        int h = l >> 3;
        float a = p * 0.17677669529663687f;   // 1/sqrt(32)
        alpha[(size_t)e * 2 + h] = a;
        atomicMaxF32(mbuf + (size_t)dst * 2 + h, a);
    }
}

// ---------------------------------------------------------------------------
// Kernel 3: ex = exp(alpha - m[dst]); accumulate denom. 1 thread / (edge,head).
// ---------------------------------------------------------------------------
__global__ __launch_bounds__(256) void edge_exp(
    const int* __restrict__ ei, float* __restrict__ alpha,
    const float* __restrict__ mbuf, float* __restrict__ denom)
{
    int t = blockIdx.x * 256 + threadIdx.x;
    if (t >= EE * 2) return;
    int e = t >> 1, h = t & 1;
    int dst = ei[EE + e];
    float ex = __expf(alpha[t] - mbuf[(size_t)dst * 2 + h]);
    alpha[t] = ex;                                   // overwrite in place
    atomicAddF32(denom + (size_t)dst * 2 + h, ex);
}

// ---------------------------------------------------------------------------
// Kernel 4: weighted scatter of values. 16 lanes/edge, float4 gather of v[src],
// 4 native f32 atomic adds per lane onto out[dst] (out holds the skip term).
// ---------------------------------------------------------------------------
__global__ __launch_bounds__(256) void edge_scatter(
    const int* __restrict__ ei, const float* __restrict__ V,
    const float* __restrict__ alpha, const float* __restrict__ denom,
    float* __restrict__ out)
{
    int t = blockIdx.x * 256 + threadIdx.x;
    int e = t >> 4;
    if (e >= EE) return;
    int l   = t & 15;
    int h   = l >> 3;
    int src = ei[e];
    int dst = ei[EE + e];

    float den = denom[(size_t)dst * 2 + h];
    float w   = alpha[(size_t)e * 2 + h] / fmaxf(den, 1e-16f);

    float4 v4 = ((const float4*)(V + (size_t)src * HC))[l];
    float* o  = out + (size_t)dst * HC + l * 4;
    atomicAddF32(o + 0, v4.x * w);
    atomicAddF32(o + 1, v4.y * w);
    atomicAddF32(o + 2, v4.z * w);
    atomicAddF32(o + 3, v4.w * w);
}

// ---------------------------------------------------------------------------
extern "C" void kernel_launch(void* const* d_in, const int* in_sizes, int n_in,
                              void* d_out, int out_size, void* d_ws, size_t ws_size,
                              hipStream_t stream)
{
    const float* x   = (const float*)d_in[0];
    const int*   ei  = (const int*)  d_in[1];
    const float* Wq  = (const float*)d_in[2];
    const float* bq  = (const float*)d_in[3];
    const float* Wk  = (const float*)d_in[4];
    const float* bk  = (const float*)d_in[5];
    const float* Wv  = (const float*)d_in[6];
    const float* bv  = (const float*)d_in[7];
    const float* Wsk = (const float*)d_in[8];
    const float* bsk = (const float*)d_in[9];
    float* out = (float*)d_out;

    // Workspace layout (floats): Q | K | V | alpha(E*2) | m(N*2) | denom(N*2)
    float* ws    = (float*)d_ws;
    float* Q     = ws;
    float* K     = ws + 3200000;    // N*HC
    float* V     = ws + 6400000;
    float* alpha = ws + 9600000;    // E*2 = 3.2M
    float* mbuf  = ws + 12800000;   // N*2
    float* denom = ws + 12900000;   // N*2

    init_md<<<(NN * 2 + 255) / 256, 256, 0, stream>>>(mbuf, denom);
    gemm_qkvs<<<NN / 16, 128, 0, stream>>>(x, Wq, bq, Wk, bk, Wv, bv, Wsk, bsk,
                                           Q, K, V, out);
    edge_alpha<<<(EE * 16 + 255) / 256, 256, 0, stream>>>(ei, Q, K, alpha, mbuf);
    edge_exp<<<(EE * 2 + 255) / 256, 256, 0, stream>>>(ei, alpha, mbuf, denom);
    edge_scatter<<<(EE * 16 + 255) / 256, 256, 0, stream>>>(ei, V, alpha, denom, out);
}